// SoftPattern_51960514347086
// MI455X (gfx1250) — compile-verified
//
#include <hip/hip_runtime.h>

// ---------------------------------------------------------------------------
// score = sum_{s=0..T-5} prod_{i=0..4} sigmoid( emb[doc[s+i]] . w[i,i+1] + b[i,i+1] )
// Kernel 1: WMMA f32 16x16x4 tiles -> vals[5][T]   (gather-bound, ~40MB traffic)
// Kernel 2: 5-window products + per-block tree reduction -> 128 partials
// Kernel 3: single-block deterministic sum -> d_out[0]
// ---------------------------------------------------------------------------

#define T_LEN   32768
#define WDIM    300
#define PCOLS   5          // superdiagonal entries (p-1)
#define TILE_M  16
#define WPB     8          // waves per block
#define BTHR    256        // block threads (8 waves of 32)

typedef __attribute__((ext_vector_type(2))) float v2f;
typedef __attribute__((ext_vector_type(8))) float v8f;

#if defined(__has_builtin)
#if __has_builtin(__builtin_amdgcn_wmma_f32_16x16x4_f32)
#define HAVE_WMMA_F32X4 1
#endif
#endif

__global__ __launch_bounds__(BTHR) void sp_vals_kernel(
    const int*   __restrict__ doc,
    const float* __restrict__ emb,   // [VOCAB][300]
    const float* __restrict__ w,     // [6][6][300]
    const float* __restrict__ b,     // [6][6]
    float*       __restrict__ vals)  // [PCOLS][T_LEN]
{
    // Stage superdiagonal weight rows into LDS, zero-padded to 16 rows so the
    // B fragment is valid for all 32 lanes (WMMA needs EXEC all-ones).
    __shared__ float wl[16 * WDIM];
    for (int e = threadIdx.x; e < 16 * WDIM; e += BTHR) {
        int n = e / WDIM;
        int k = e - n * WDIM;
        wl[e] = (n < PCOLS) ? w[(7 * n + 1) * WDIM + k] : 0.0f;  // w[n][n+1][k]
    }
    __syncthreads();

    const int wave = threadIdx.x >> 5;
    const int lane = threadIdx.x & 31;
    const int ncol = lane & 15;   // N for B/C/D; M row this lane feeds for A
    const int half = lane >> 4;   // K-half select for A/B fragments
    const int tileBase = (blockIdx.x * WPB + wave) * TILE_M;

    // A-fragment source: embedding row of token (tileBase + ncol); lane
    // supplies K = kb, kb+1 per step (8-byte aligned: 300*4 % 8 == 0).
    const int tokA = tileBase + ncol;
    const float* __restrict__ xrow = emb + (size_t)doc[tokA] * WDIM;
    const float* __restrict__ wrow = &wl[ncol * WDIM];
    const int kb = half * 2;

    v8f c = {0.f, 0.f, 0.f, 0.f, 0.f, 0.f, 0.f, 0.f};

#if HAVE_WMMA_F32X4
    for (int k0 = 0; k0 < WDIM; k0 += 4) {
        v2f a  = *(const v2f*)(xrow + k0 + kb);   // A[M=ncol][k0+kb .. +1]
        v2f bb = *(const v2f*)(wrow + k0 + kb);   // B[k0+kb .. +1][N=ncol]
        c = __builtin_amdgcn_wmma_f32_16x16x4_f32(
                /*neg_a=*/false, a, /*neg_b=*/false, bb,
                /*c_mod=*/(short)0, c, /*reuse_a=*/false, /*reuse_b=*/false);
    }
#else
    // Scalar fallback matching the C/D layout: c[r] = x_{tileBase+r+8*half} . w[ncol]
    for (int r = 0; r < 8; ++r) {
        const int tokM = tileBase + r + 8 * half;
        const float* xr = emb + (size_t)doc[tokM] * WDIM;
        float acc = 0.f;
        for (int k = 0; k < WDIM; ++k) acc += xr[k] * wrow[k];
        c[r] = acc;
    }
#endif

    // C/D layout: VGPR r -> M = r + 8*half, N = ncol. Only N<5 is real.
    if (ncol < PCOLS) {
        const float bn = b[7 * ncol + 1];
#pragma unroll
        for (int r = 0; r < 8; ++r) {
            const int m = r + 8 * half;
            const float z = c[r] + bn;
            const float s = 1.0f / (1.0f + __expf(-z));
            vals[ncol * T_LEN + tileBase + m] = s;
        }
    }
}

__global__ __launch_bounds__(BTHR) void sp_window_kernel(
    const float* __restrict__ vals, float* __restrict__ partials)
{
    const int s = blockIdx.x * BTHR + threadIdx.x;
    float p = 0.f;
    if (s <= T_LEN - PCOLS) {
        p = vals[0 * T_LEN + s + 0] *
            vals[1 * T_LEN + s + 1] *
            vals[2 * T_LEN + s + 2] *
            vals[3 * T_LEN + s + 3] *
            vals[4 * T_LEN + s + 4];
    }
    __shared__ float red[BTHR];
    red[threadIdx.x] = p;
    __syncthreads();
    for (int off = BTHR / 2; off > 0; off >>= 1) {
        if (threadIdx.x < off) red[threadIdx.x] += red[threadIdx.x + off];
        __syncthreads();
    }
    if (threadIdx.x == 0) partials[blockIdx.x] = red[0];
}

__global__ __launch_bounds__(128) void sp_final_kernel(
    const float* __restrict__ partials, float* __restrict__ out)
{
    __shared__ float red[128];
    red[threadIdx.x] = partials[threadIdx.x];
    __syncthreads();
    for (int off = 64; off > 0; off >>= 1) {
        if (threadIdx.x < off) red[threadIdx.x] += red[threadIdx.x + off];
        __syncthreads();
    }
    if (threadIdx.x == 0) out[0] = red[0];
}

extern "C" void kernel_launch(void* const* d_in, const int* in_sizes, int n_in,
                              void* d_out, int out_size, void* d_ws, size_t ws_size,
                              hipStream_t stream) {
    const int*   doc = (const int*)d_in[0];
    const float* emb = (const float*)d_in[1];
    const float* w   = (const float*)d_in[2];
    const float* b   = (const float*)d_in[3];

    float* vals     = (float*)d_ws;                       // 5*T floats = 640 KB
    float* partials = vals + (size_t)PCOLS * T_LEN;       // 128 floats

    const int nblk1 = T_LEN / (TILE_M * WPB);             // 256 blocks, 1 tile/wave
    sp_vals_kernel<<<nblk1, BTHR, 0, stream>>>(doc, emb, w, b, vals);

    const int nblk2 = T_LEN / BTHR;                       // 128 blocks
    sp_window_kernel<<<nblk2, BTHR, 0, stream>>>(vals, partials);

    sp_final_kernel<<<1, 128, 0, stream>>>(partials, (float*)d_out);
}